// VariationalFFNGradientEngine_12996571038352
// MI455X (gfx1250) — compile-verified
//
#include <hip/hip_runtime.h>
#include <stdint.h>

// Problem constants (match reference)
#define BB 8
#define NNN 128
#define KD 16
#define ALPHA_C 0.01f
#define LAM_C 1.0f
#define KAPPA_C 1.0f
#define EPS_C 1e-6f
#define BN (BB * NNN)        // 1024
#define BNK (BN * KD)        // 16384

typedef __attribute__((ext_vector_type(2))) float v2f;
typedef __attribute__((ext_vector_type(8))) float v8f;

// Workspace offsets (floats). Total 1,508,352 floats = ~6.03 MB.
#define OFF_E   0u          // (BN,256)  E_n row-major
#define OFF_A   262144u     // (BN,256)  A_n = E^T diag(r) E
#define OFF_Q   524288u     // (BN,256)  vec(u u^T)
#define OFF_U   786432u     // (BN,16)   u = E^T mu_q
#define OFF_C   802816u     // (BN,16)   c = A u
#define OFF_D   819200u     // (BN)      d = u.c
#define OFF_T   820224u     // (B,N,N)   t = kl * beta
#define OFF_S   951296u     // (BN,256)  S   = beta @ A
#define OFF_SS  1213440u    // (BN,256)  Ssm = t @ A
#define OFF_BC  1475584u    // (BN,16)   beta @ c
#define OFF_TC  1491968u    // (BN,16)   t @ c

// ---------------------------------------------------------------------------
// CDNA5 async global->LDS staging (ASYNCcnt path, §15.18.3 opcode 98).
// VDST = LDS byte address VGPR, VADDR = 64-bit global address.
// ---------------------------------------------------------------------------
__device__ __forceinline__ void async_load_b128(uint32_t lds_off, const void* gaddr) {
  asm volatile("global_load_async_to_lds_b128 %0, %1, off"
               :: "v"(lds_off), "v"((unsigned long long)(uintptr_t)gaddr)
               : "memory");
}
__device__ __forceinline__ void wait_async0() {
  asm volatile("s_wait_asynccnt 0x0" ::: "memory");
}

// ---------------------------------------------------------------------------
// Wave-level 16x16 f32 GEMM tile accumulate using V_WMMA_F32_16X16X4_F32.
//   D[m][n] += sum_k A[m*a_rs + k*a_ks] * B[k*b_ks + n*b_ns]
// ISA layouts (wave32):
//   A 16x4 : lane L holds A[L&15][k0 + 2*(L>>4) + {0,1}] in 2 VGPRs
//   B  4x16: lane L holds B[k0 + 2*(L>>4) + {0,1}][L&15]
//   C/D    : vgpr v in lane L = D[v + 8*(L>>4)][L&15]
// ---------------------------------------------------------------------------
__device__ __forceinline__ v8f wmma_acc(const float* __restrict__ A, int a_rs, int a_ks,
                                        const float* __restrict__ Bm, int b_ks, int b_ns,
                                        int Klen, v8f acc) {
  const int lane = threadIdx.x & 31;
  const int half = lane >> 4;
  const int l16  = lane & 15;
  for (int k0 = 0; k0 < Klen; k0 += 4) {
    const int ka = k0 + 2 * half;
    v2f a, b;
    a[0] = A[l16 * a_rs + (ka + 0) * a_ks];
    a[1] = A[l16 * a_rs + (ka + 1) * a_ks];
    b[0] = Bm[(ka + 0) * b_ks + l16 * b_ns];
    b[1] = Bm[(ka + 1) * b_ks + l16 * b_ns];
    acc = __builtin_amdgcn_wmma_f32_16x16x4_f32(false, a, false, b,
                                                (short)0, acc, false, false);
  }
  return acc;
}

// Fragment-based variant: A fragments pre-loaded in registers (K4 k-steps of 4).
template <int K4>
__device__ __forceinline__ v8f wmma_accB(const v2f* afrag,
                                         const float* __restrict__ Bm,
                                         int b_ks, int b_ns, v8f acc) {
  const int lane = threadIdx.x & 31;
  const int half = lane >> 4;
  const int l16  = lane & 15;
#pragma unroll
  for (int k4 = 0; k4 < K4; ++k4) {
    const int ka = k4 * 4 + 2 * half;
    v2f b;
    b[0] = Bm[(ka + 0) * b_ks + l16 * b_ns];
    b[1] = Bm[(ka + 1) * b_ks + l16 * b_ns];
    acc = __builtin_amdgcn_wmma_f32_16x16x4_f32(false, afrag[k4], false, b,
                                                (short)0, acc, false, false);
  }
  return acc;
}

// ---------------------------------------------------------------------------
// K1: per (b,n) node setup. 1 wave per node.
//   E = expm(phi.G) via scaling(2^-6)+Taylor(8)+6 squarings in LDS
//   u = E^T mu_q ; A = E^T diag(1/(clip(sq)+eps)) E ; c = A u ; d = u.c
//   Q = vec(u u^T) ; grad_sigma written directly to out[BNK..)
// ---------------------------------------------------------------------------
__global__ void prep_kernel(const float* __restrict__ mu_q,
                            const float* __restrict__ sigma_q,
                            const float* __restrict__ sigma_p,
                            const float* __restrict__ phi,
                            const float* __restrict__ gen,
                            float* __restrict__ ws,
                            float* __restrict__ out) {
  __shared__ float Xs[256];
  __shared__ float Eb[2][256];
  __shared__ float Tb[2][256];
  __shared__ float su[16], sr[16], sc2[16];

  const int bi = blockIdx.x;            // 0..BN-1
  const int lane = threadIdx.x;

  const float p0 = phi[bi * 3 + 0];
  const float p1 = phi[bi * 3 + 1];
  const float p2 = phi[bi * 3 + 2];
  const float sc = 1.0f / 64.0f;        // 2^-6 scaling

  for (int e = lane; e < 256; e += 32)
    Xs[e] = (p0 * gen[e] + p1 * gen[256 + e] + p2 * gen[512 + e]) * sc;
  __syncthreads();

  for (int e = lane; e < 256; e += 32) {
    const float id = ((e >> 4) == (e & 15)) ? 1.0f : 0.0f;
    Eb[0][e] = id + Xs[e];
    Tb[0][e] = Xs[e];
  }
  __syncthreads();

  int et = 0, tt = 0;
  for (int k = 2; k <= 8; ++k) {        // Taylor terms
    const float rk = 1.0f / (float)k;
    for (int e = lane; e < 256; e += 32) {
      const int r = e >> 4, cc = e & 15;
      float a = 0.0f;
      for (int m = 0; m < 16; ++m) a += Tb[tt][r * 16 + m] * Xs[m * 16 + cc];
      Tb[tt ^ 1][e] = a * rk;
    }
    __syncthreads();
    tt ^= 1;
    for (int e = lane; e < 256; e += 32) Eb[et][e] += Tb[tt][e];
    __syncthreads();
  }
  for (int s = 0; s < 6; ++s) {         // squarings
    for (int e = lane; e < 256; e += 32) {
      const int r = e >> 4, cc = e & 15;
      float a = 0.0f;
      for (int m = 0; m < 16; ++m) a += Eb[et][r * 16 + m] * Eb[et][m * 16 + cc];
      Eb[et ^ 1][e] = a;
    }
    __syncthreads();
    et ^= 1;
  }
  const float* E = Eb[et];

  if (lane < 16) {
    float acc = 0.0f;                   // u[k] = (E^T mu)[k]
    for (int m = 0; m < 16; ++m) acc += E[m * 16 + lane] * mu_q[bi * 16 + m];
    su[lane] = acc;
    const float sq = fmaxf(sigma_q[bi * 16 + lane], EPS_C);
    sr[lane] = 1.0f / (sq + EPS_C);
    const float sp = fmaxf(sigma_p[bi * 16 + lane], EPS_C);
    out[BNK + bi * 16 + lane] = ALPHA_C * 0.5f * (1.0f / sp - 1.0f / sq);
  }
  __syncthreads();

  for (int e = lane; e < 256; e += 32) {
    const int k = e >> 4, l = e & 15;
    float a = 0.0f;                     // A[k][l] = sum_m E[m][k] r[m] E[m][l]
    for (int m = 0; m < 16; ++m) a += E[m * 16 + k] * sr[m] * E[m * 16 + l];
    ws[OFF_A + bi * 256 + e] = a;
    ws[OFF_Q + bi * 256 + e] = su[k] * su[l];
    ws[OFF_E + bi * 256 + e] = E[e];
    Tb[0][e] = a;                       // keep A in LDS for c = A u
  }
  __syncthreads();

  if (lane < 16) {
    float c = 0.0f;
    for (int l = 0; l < 16; ++l) c += Tb[0][lane * 16 + l] * su[l];
    sc2[lane] = c;
    ws[OFF_C + bi * 16 + lane] = c;
    ws[OFF_U + bi * 16 + lane] = su[lane];
  }
  __syncthreads();
  if (lane == 0) {
    float d = 0.0f;
    for (int k = 0; k < 16; ++k) d += su[k] * sc2[k];
    ws[OFF_D + bi] = d;
  }
}

// ---------------------------------------------------------------------------
// K2: kl matrix + t = kl*beta.  kl = 0.5*(Q A^T) - (U C^T) + 0.5 d_j
// grid = B*8 (b, ti) blocks x 256 threads (8 waves, one tj tile per wave).
// The 16x256 Q row-panel (16 KB, contiguous) is staged once into LDS via
// async b128 copies and shared by all 8 waves. No in-wave reuse loop ->
// nothing for LICM to hoist -> no spills; k-steps pipeline ds + global loads.
// ---------------------------------------------------------------------------
__global__ void __launch_bounds__(256, 1)
klt_kernel(const float* __restrict__ beta,
           const float* __restrict__ ws,
           float* __restrict__ ws_t) {
  __shared__ float shQ[16 * 256];       // 16 KB
  const int b  = blockIdx.x >> 3;
  const int ti = blockIdx.x & 7;
  const int t  = threadIdx.x;
  const int tj = t >> 5;                // wave id = output tj tile
  const int lane = t & 31;
  const int half = lane >> 4, l16 = lane & 15;

  const char* Qg = (const char*)(ws + OFF_Q + (b * NNN + ti * 16) * 256);
  const uint32_t shQ_base = (uint32_t)(uintptr_t)(void*)shQ;
#pragma unroll
  for (int i = 0; i < 4; ++i) {         // 1024 x 16B chunks, 4 per thread
    const uint32_t cidx = (uint32_t)(i * 256 + t);
    async_load_b128(shQ_base + cidx * 16u, Qg + (size_t)cidx * 16u);
  }
  wait_async0();
  __syncthreads();

  const float* Afl = ws + OFF_A + (b * NNN + tj * 16) * 256;
  const float* U   = ws + OFF_U + (b * NNN + ti * 16) * 16;
  const float* C   = ws + OFF_C + (b * NNN + tj * 16) * 16;
  const float* dv  = ws + OFF_D + b * NNN + tj * 16;

  v8f accQA = {0.f, 0.f, 0.f, 0.f, 0.f, 0.f, 0.f, 0.f};
  v8f accUC = {0.f, 0.f, 0.f, 0.f, 0.f, 0.f, 0.f, 0.f};
  // D += Q(16x256, LDS) * Afl^T : B[k][j] = Afl[j*256 + k]
  accQA = wmma_acc(shQ, 256, 1, Afl, 1, 256, 256, accQA);
  // D += U(16x16) * C^T : B[k][j] = C[j*16 + k]
  accUC = wmma_acc(U, 16, 1, C, 1, 16, 16, accUC);

  for (int v = 0; v < 8; ++v) {
    const int i = ti * 16 + v + 8 * half;
    const int j = tj * 16 + l16;
    const float kl = 0.5f * accQA[v] - accUC[v] + 0.5f * dv[l16];
    const int idx = (b * NNN + i) * NNN + j;
    ws_t[idx] = kl * beta[idx];
  }
}

// ---------------------------------------------------------------------------
// K3: reductions over j: {S,bc} = beta @ [A|c], {Ssm,tc} = t @ [A|c]
// grid = B*2*8 (b, w, ti) blocks x 256 threads (8 waves).
// The 16x128 W row-panel (8 KB, contiguous) is staged into LDS via async
// b128 copies; each wave then pre-loads its A fragments into 32 v2f regs
// (explicit, bounded reuse) and sweeps its subset of the 17 column tiles.
// ---------------------------------------------------------------------------
__global__ void __launch_bounds__(256, 1)
reduce_kernel(const float* __restrict__ beta,
              const float* __restrict__ ws_in,
              float* __restrict__ ws_out) {
  __shared__ float shW[16 * 128];       // 8 KB
  int idx = blockIdx.x;
  const int ti = idx & 7;  idx >>= 3;
  const int w  = idx & 1;
  const int b  = idx >> 1;
  const int t  = threadIdx.x;
  const int wv = t >> 5;
  const int lane = t & 31;
  const int half = lane >> 4, l16 = lane & 15;

  const float* W = (w == 0) ? (beta + b * NNN * NNN)
                            : (ws_in + OFF_T + b * NNN * NNN);
  const char* Wg = (const char*)(W + ti * 16 * NNN);
  const uint32_t shW_base = (uint32_t)(uintptr_t)(void*)shW;
#pragma unroll
  for (int i = 0; i < 2; ++i) {         // 512 x 16B chunks, 2 per thread
    const uint32_t cidx = (uint32_t)(i * 256 + t);
    async_load_b128(shW_base + cidx * 16u, Wg + (size_t)cidx * 16u);
  }
  wait_async0();
  __syncthreads();

  // Explicit A-fragment preload: 32 k-steps x v2f = 64 VGPRs, reused by all
  // ct tiles handled by this wave.
  v2f afrag[32];
#pragma unroll
  for (int k4 = 0; k4 < 32; ++k4) {
    const int ka = k4 * 4 + 2 * half;
    afrag[k4][0] = shW[l16 * 128 + ka];
    afrag[k4][1] = shW[l16 * 128 + ka + 1];
  }

  for (int ct = wv; ct < 17; ct += 8) {
    v8f acc = {0.f, 0.f, 0.f, 0.f, 0.f, 0.f, 0.f, 0.f};
    if (ct < 16) {
      const float* Bm = ws_in + OFF_A + b * NNN * 256 + ct * 16; // B[j][n]=base[j*256+n]
      acc = wmma_accB<32>(afrag, Bm, 256, 1, acc);
      float* outp = ws_out + ((w == 0) ? OFF_S : OFF_SS);
      for (int v = 0; v < 8; ++v) {
        const int i = ti * 16 + v + 8 * half;
        outp[(b * NNN + i) * 256 + ct * 16 + l16] = acc[v];
      }
    } else {
      const float* Bm = ws_in + OFF_C + b * NNN * 16;            // B[j][n]=base[j*16+n]
      acc = wmma_accB<32>(afrag, Bm, 16, 1, acc);
      float* outp = ws_out + ((w == 0) ? OFF_BC : OFF_TC);
      for (int v = 0; v < 8; ++v) {
        const int i = ti * 16 + v + 8 * half;
        outp[(b * NNN + i) * 16 + l16] = acc[v];
      }
    }
  }
}

// ---------------------------------------------------------------------------
// K4: per (b,i) finish:
//   P = S_i u_i - bc_i ; T = Ssm_i u_i - tc_i ; s_i = sum_j t_ij
//   grad_mu = ALPHA*(mu_q-mu_p)/sp + E_i * (LAM*P + (LAM/KAPPA)*(T - s_i*P))
// ---------------------------------------------------------------------------
__global__ void final_kernel(const float* __restrict__ mu_q,
                             const float* __restrict__ mu_p,
                             const float* __restrict__ sigma_p,
                             const float* __restrict__ ws,
                             float* __restrict__ out) {
  __shared__ float shu[16], shhat[16], shred[32];
  const int bi = blockIdx.x;
  const int lane = threadIdx.x;

  if (lane < 16) shu[lane] = ws[OFF_U + bi * 16 + lane];

  float part = 0.0f;
  for (int j = lane; j < NNN; j += 32) part += ws[OFF_T + (size_t)bi * NNN + j];
  shred[lane] = part;
  __syncthreads();
  if (lane == 0) {
    float s = 0.0f;
    for (int q = 0; q < 32; ++q) s += shred[q];
    shred[0] = s;
  }
  __syncthreads();
  const float s_i = shred[0];

  if (lane < 16) {
    const float* Sr  = ws + OFF_S  + bi * 256 + lane * 16;
    const float* Smr = ws + OFF_SS + bi * 256 + lane * 16;
    float P = 0.0f, T = 0.0f;
    for (int l = 0; l < 16; ++l) { P += Sr[l] * shu[l]; T += Smr[l] * shu[l]; }
    P -= ws[OFF_BC + bi * 16 + lane];
    T -= ws[OFF_TC + bi * 16 + lane];
    shhat[lane] = LAM_C * P + (LAM_C / KAPPA_C) * (T - s_i * P);
  }
  __syncthreads();

  if (lane < 16) {
    const float* E = ws + OFF_E + bi * 256 + lane * 16;   // row k of E_i
    float acc = 0.0f;
    for (int l = 0; l < 16; ++l) acc += E[l] * shhat[l];
    const float sp = fmaxf(sigma_p[bi * 16 + lane], EPS_C);
    const float self = ALPHA_C * (mu_q[bi * 16 + lane] - mu_p[bi * 16 + lane]) / sp;
    out[bi * 16 + lane] = self + acc;
  }
}

// ---------------------------------------------------------------------------
extern "C" void kernel_launch(void* const* d_in, const int* in_sizes, int n_in,
                              void* d_out, int out_size, void* d_ws, size_t ws_size,
                              hipStream_t stream) {
  (void)in_sizes; (void)n_in; (void)out_size; (void)ws_size;
  const float* mu_q    = (const float*)d_in[0];
  const float* sigma_q = (const float*)d_in[1];
  const float* mu_p    = (const float*)d_in[2];
  const float* sigma_p = (const float*)d_in[3];
  const float* beta    = (const float*)d_in[4];
  const float* phi     = (const float*)d_in[5];
  const float* gen     = (const float*)d_in[6];
  float* out = (float*)d_out;
  float* ws  = (float*)d_ws;

  prep_kernel<<<BN, 32, 0, stream>>>(mu_q, sigma_q, sigma_p, phi, gen, ws, out);
  klt_kernel<<<BB * 8, 256, 0, stream>>>(beta, ws, ws + OFF_T);
  reduce_kernel<<<BB * 2 * 8, 256, 0, stream>>>(beta, ws, ws);
  final_kernel<<<BN, 32, 0, stream>>>(mu_q, mu_p, sigma_p, ws, out);
}